// GATGraphClassifier_15126874816619
// MI455X (gfx1250) — compile-verified
//
#include <hip/hip_runtime.h>
#include <math.h>

// ---------------------------------------------------------------------------
// GAT graph classifier for MI455X (gfx1250).
// Dense projections use native fp32 WMMA (v_wmma_f32_16x16x4_f32), with both
// operands staged through LDS via coalesced 128-bit global loads so the WMMA
// main loop is branch-free. Sparse softmax/aggregation uses global f32
// atomics (node arrays are 25.6MB -> resident in the 192MB L2).
// ---------------------------------------------------------------------------

typedef float v2f __attribute__((ext_vector_type(2)));
typedef float v8f __attribute__((ext_vector_type(8)));

#define HDIM 64
#define OUTC 10
#define LSTR 65   // padded LDS row stride (floats) to kill bank conflicts

// ---------------- utility -----------------------------------------------

__global__ void fill_f32(float* __restrict__ p, float v, int n) {
    int i = blockIdx.x * blockDim.x + threadIdx.x;
    if (i < n) p[i] = v;
}

// monotone float->uint key for atomicMax on floats
__device__ __forceinline__ unsigned fkey(float f) {
    unsigned u = __float_as_uint(f);
    return (u & 0x80000000u) ? ~u : (u | 0x80000000u);
}
__device__ __forceinline__ float funkey(unsigned k) {
    unsigned u = (k & 0x80000000u) ? (k & 0x7FFFFFFFu) : ~k;
    return __uint_as_float(u);
}

// we_dot = sum_h We[h] * a_edge[h]   (edge_dim == 1)
__global__ void wedot_kernel(const float* __restrict__ We,
                             const float* __restrict__ a_edge,
                             float* __restrict__ wd) {
    if (threadIdx.x == 0 && blockIdx.x == 0) {
        float s = 0.f;
        for (int h = 0; h < HDIM; ++h) s += We[h] * a_edge[h];
        wd[0] = s;
    }
}

// self-loop attr: s[dst] += attr, cnt[dst] += 1
__global__ void loop_attr_kernel(const int* __restrict__ ei,
                                 const float* __restrict__ ea,
                                 float* __restrict__ sloop,
                                 float* __restrict__ cnt, int E) {
    int i = blockIdx.x * blockDim.x + threadIdx.x;
    if (i >= E) return;
    int d = ei[E + i];
    atomicAdd(&sloop[d], ea[i]);
    atomicAdd(&cnt[d], 1.0f);
}

// ---------------- WMMA GEMM:  Y[n][h] = sum_k X[n][k] * W[h][k] ----------
// X: nrows x 64 row-major, W: 64 x 64 row-major (H x K), Y: nrows x 64.
// 128 threads = 4 waves; wave w owns rows [blk*64 + w*16, +16), all 64 cols.
// W and the 64-row X tile are staged in padded LDS with coalesced b128 loads;
// the WMMA loop is then branch-free (ds loads + v_wmma only).
__global__ __launch_bounds__(128)
void gemm_wmma_f32(const float* __restrict__ X, const float* __restrict__ W,
                   float* __restrict__ Y, int nrows) {
    __shared__ float lw[HDIM * LSTR];
    __shared__ float lx[HDIM * LSTR];
    int tid = threadIdx.x;
    long rowBase = (long)blockIdx.x * 64;

    // stage W: 1024 float4s, 8 per thread, fully coalesced
    for (int i = tid; i < HDIM * HDIM / 4; i += 128) {
        int h = i >> 4;           // row of W (output channel)
        int k = (i & 15) * 4;     // col of W (input channel)
        float4 w4 = ((const float4*)W)[i];
        float* p = &lw[h * LSTR + k];
        p[0] = w4.x; p[1] = w4.y; p[2] = w4.z; p[3] = w4.w;
    }
    // stage X tile (64 rows), OOB rows zero-filled once here
    for (int i = tid; i < HDIM * HDIM / 4; i += 128) {
        int r = i >> 4;
        int c = (i & 15) * 4;
        long row = rowBase + r;
        float4 x4 = make_float4(0.f, 0.f, 0.f, 0.f);
        if (row < nrows) x4 = *(const float4*)(X + row * HDIM + c);
        float* p = &lx[r * LSTR + c];
        p[0] = x4.x; p[1] = x4.y; p[2] = x4.z; p[3] = x4.w;
    }
    __syncthreads();

    int wave  = tid >> 5;
    int lane  = tid & 31;
    int m16   = lane & 15;      // row within 16x16 tile / B column idx
    int khalf = lane >> 4;      // K-pair select for f32 fragments

    v8f acc[4] = {v8f{}, v8f{}, v8f{}, v8f{}};

    for (int k0 = 0; k0 < HDIM; k0 += 4) {
        // A fragment: lane holds X[wave*16+m16][k0 + khalf*2 + {0,1}]
        const float* axp = &lx[(wave * 16 + m16) * LSTR + k0 + khalf * 2];
        v2f a;
        a.x = axp[0];
        a.y = axp[1];
#pragma unroll
        for (int t = 0; t < 4; ++t) {
            // B fragment: B[k][n] = W[n][k] with n = t*16 + m16
            const float* bp = &lw[(t * 16 + m16) * LSTR + k0 + khalf * 2];
            v2f b;
            b.x = bp[0];
            b.y = bp[1];
            acc[t] = __builtin_amdgcn_wmma_f32_16x16x4_f32(
                false, a, false, b, (short)0, acc[t], false, false);
        }
    }

    // C/D layout: VGPR v, lanes 0-15 -> M=v, lanes 16-31 -> M=v+8, N=lane%16
    long tileM = rowBase + wave * 16;
    int mbase = (lane >> 4) * 8;
#pragma unroll
    for (int t = 0; t < 4; ++t) {
#pragma unroll
        for (int v = 0; v < 8; ++v) {
            long row = tileM + mbase + v;
            if (row < nrows) Y[(size_t)row * HDIM + t * 16 + m16] = acc[t][v];
        }
    }
}

// ---------------- per-node attention scores ------------------------------
// wave per node: asrc[n]=xp[n]·a_src, adst[n]=xp[n]·a_dst; init mkey/denom.
__global__ void node_scores_kernel(const float* __restrict__ xp,
                                   const float* __restrict__ a_src,
                                   const float* __restrict__ a_dst,
                                   float* __restrict__ asrc,
                                   float* __restrict__ adst,
                                   unsigned* __restrict__ mkey,
                                   float* __restrict__ denom, int N) {
    int lane = threadIdx.x & 31;
    int n = blockIdx.x * (blockDim.x >> 5) + (threadIdx.x >> 5);
    if (n >= N) return;
    float x0 = xp[(size_t)n * HDIM + lane];
    float x1 = xp[(size_t)n * HDIM + 32 + lane];
    float ps = x0 * a_src[lane] + x1 * a_src[lane + 32];
    float pd = x0 * a_dst[lane] + x1 * a_dst[lane + 32];
    for (int off = 16; off > 0; off >>= 1) {
        ps += __shfl_down(ps, off);
        pd += __shfl_down(pd, off);
    }
    if (lane == 0) {
        asrc[n] = ps;
        adst[n] = pd;
        mkey[n] = 0u;       // below any real fkey()
        denom[n] = 0.f;
    }
}

// ---------------- edge pass 1: e + segment max ---------------------------
// edge ids [0,E) = real edges, [E, E+N) = self loops with mean edge attr.
__global__ void edge_max_kernel(const int* __restrict__ ei,
                                const float* __restrict__ ea,
                                const float* __restrict__ sloop,
                                const float* __restrict__ cnt,
                                const float* __restrict__ asrc,
                                const float* __restrict__ adst,
                                const float* __restrict__ wd,
                                float* __restrict__ pbuf,
                                unsigned* __restrict__ mkey,
                                int E, int N) {
    int i = blockIdx.x * blockDim.x + threadIdx.x;
    if (i >= E + N) return;
    int s, d; float attr;
    if (i < E) { s = ei[i]; d = ei[E + i]; attr = ea[i]; }
    else { s = i - E; d = s; attr = sloop[s] / fmaxf(cnt[s], 1.0f); }
    float e = asrc[s] + adst[d] + attr * wd[0];
    e = (e > 0.f) ? e : 0.2f * e;              // leaky_relu(0.2)
    pbuf[i] = e;
    atomicMax(&mkey[d], fkey(e));
}

// ---------------- edge pass 2: exp + segment denom -----------------------
__global__ void edge_exp_kernel(const int* __restrict__ ei,
                                const unsigned* __restrict__ mkey,
                                float* __restrict__ pbuf,
                                float* __restrict__ denom, int E, int N) {
    int i = blockIdx.x * blockDim.x + threadIdx.x;
    if (i >= E + N) return;
    int d = (i < E) ? ei[E + i] : (i - E);
    float pe = __expf(pbuf[i] - funkey(mkey[d]));
    pbuf[i] = pe;
    atomicAdd(&denom[d], pe);
}

// ---------------- edge pass 3: aggregate out[dst] += alpha * xp[src] -----
// wave per edge; 2 channels per lane -> coalesced 128B gathers.
__global__ void aggregate_kernel(const int* __restrict__ ei,
                                 const float* __restrict__ xp,
                                 const float* __restrict__ pbuf,
                                 const float* __restrict__ denom,
                                 float* __restrict__ out, int E, int N) {
    int lane = threadIdx.x & 31;
    int i = blockIdx.x * (blockDim.x >> 5) + (threadIdx.x >> 5);
    if (i >= E + N) return;
    int s, d;
    if (i < E) { s = ei[i]; d = ei[E + i]; }
    else { s = i - E; d = s; }
    float alpha = pbuf[i] / denom[d];
    float v0 = xp[(size_t)s * HDIM + lane];
    float v1 = xp[(size_t)s * HDIM + 32 + lane];
    atomicAdd(&out[(size_t)d * HDIM + lane], alpha * v0);
    atomicAdd(&out[(size_t)d * HDIM + 32 + lane], alpha * v1);
}

// ---------------- bias (+ optional relu) ---------------------------------
__global__ void bias_act_kernel(float* __restrict__ h, const float* __restrict__ b,
                                int total, int relu) {
    int i = blockIdx.x * blockDim.x + threadIdx.x;
    if (i >= total) return;
    float v = h[i] + b[i & (HDIM - 1)];
    h[i] = relu ? fmaxf(v, 0.f) : v;
}

// ---------------- mean pool over graphs ----------------------------------
__global__ void pool_kernel(const float* __restrict__ h, const int* __restrict__ batch,
                            float* __restrict__ pooled, float* __restrict__ gcnt, int N) {
    int lane = threadIdx.x & 31;
    int n = blockIdx.x * (blockDim.x >> 5) + (threadIdx.x >> 5);
    if (n >= N) return;
    int g = batch[n];
    atomicAdd(&pooled[(size_t)g * HDIM + lane],      h[(size_t)n * HDIM + lane]);
    atomicAdd(&pooled[(size_t)g * HDIM + 32 + lane], h[(size_t)n * HDIM + 32 + lane]);
    if (lane == 0) atomicAdd(&gcnt[g], 1.0f);
}

// ---------------- FC head + log_softmax (one wave per graph) -------------
__global__ void head_kernel(const float* __restrict__ pooled,
                            const float* __restrict__ gcnt,
                            const float* __restrict__ fcW,
                            const float* __restrict__ fcb,
                            float* __restrict__ out, int G) {
    int g = blockIdx.x;
    if (g >= G) return;
    int lane = threadIdx.x & 31;
    float c = fmaxf(gcnt[g], 1.0f);
    float h0 = pooled[(size_t)g * HDIM + lane] / c;
    float h1 = pooled[(size_t)g * HDIM + 32 + lane] / c;
    float logit[OUTC];
#pragma unroll
    for (int o = 0; o < OUTC; ++o) {
        float p = h0 * fcW[o * HDIM + lane] + h1 * fcW[o * HDIM + 32 + lane];
        for (int off = 16; off > 0; off >>= 1) p += __shfl_down(p, off);
        logit[o] = p + fcb[o];   // valid on lane 0
    }
    if (lane == 0) {
        float mx = -INFINITY;
#pragma unroll
        for (int o = 0; o < OUTC; ++o) mx = fmaxf(mx, logit[o]);
        float se = 0.f;
#pragma unroll
        for (int o = 0; o < OUTC; ++o) se += __expf(logit[o] - mx);
        float lse = mx + __logf(se);
#pragma unroll
        for (int o = 0; o < OUTC; ++o) out[(size_t)g * OUTC + o] = logit[o] - lse;
    }
}

// ---------------------------------------------------------------------------

extern "C" void kernel_launch(void* const* d_in, const int* in_sizes, int n_in,
                              void* d_out, int out_size, void* d_ws, size_t ws_size,
                              hipStream_t stream) {
    const float* x     = (const float*)d_in[0];
    const int*   ei    = (const int*)  d_in[1];
    const float* ea    = (const float*)d_in[2];
    const int*   batch = (const int*)  d_in[3];
    const float* W[2]      = {(const float*)d_in[4],  (const float*)d_in[10]};
    const float* a_src[2]  = {(const float*)d_in[5],  (const float*)d_in[11]};
    const float* a_dst[2]  = {(const float*)d_in[6],  (const float*)d_in[12]};
    const float* We[2]     = {(const float*)d_in[7],  (const float*)d_in[13]};
    const float* a_edge[2] = {(const float*)d_in[8],  (const float*)d_in[14]};
    const float* bias[2]   = {(const float*)d_in[9],  (const float*)d_in[15]};
    const float* fcW = (const float*)d_in[16];
    const float* fcb = (const float*)d_in[17];
    float* out = (float*)d_out;

    const int N  = in_sizes[3];        // batch vector length
    const int E  = in_sizes[2];        // edge_attr length (edge_dim=1)
    const int G  = out_size / OUTC;
    const size_t N64 = (size_t)N * HDIM;

    // workspace carve-up (floats)
    float*    bufA   = (float*)d_ws;           // N*64  : xp (projected feats)
    float*    bufB   = bufA + N64;             // N*64  : aggregation / activations
    float*    sloop  = bufB + N64;             // N
    float*    cnt    = sloop + N;              // N
    float*    asrc   = cnt + N;                // N
    float*    adst   = asrc + N;               // N
    unsigned* mkey   = (unsigned*)(adst + N);  // N
    float*    denom  = (float*)(mkey + N);     // N
    float*    pbuf   = denom + N;              // E+N
    float*    pooled = pbuf + (E + N);         // G*64
    float*    gcnt   = pooled + (size_t)G * HDIM; // G
    float*    wd     = gcnt + G;               // 1

    const int T = 256;
    const int wavesPerBlk = T / 32;
    auto cdiv = [](long a, long b) { return (int)((a + b - 1) / b); };

    // one-time: self-loop mean edge attr, pooled accumulators
    fill_f32<<<cdiv(N, T), T, 0, stream>>>(sloop, 0.f, N);
    fill_f32<<<cdiv(N, T), T, 0, stream>>>(cnt, 0.f, N);
    fill_f32<<<cdiv((long)G * HDIM, T), T, 0, stream>>>(pooled, 0.f, G * HDIM);
    fill_f32<<<cdiv(G, T), T, 0, stream>>>(gcnt, 0.f, G);
    loop_attr_kernel<<<cdiv(E, T), T, 0, stream>>>(ei, ea, sloop, cnt, E);

    for (int l = 0; l < 2; ++l) {
        const float* Xin = (l == 0) ? x : bufB;
        wedot_kernel<<<1, 1, 0, stream>>>(We[l], a_edge[l], wd);
        gemm_wmma_f32<<<cdiv(N, 64), 128, 0, stream>>>(Xin, W[l], bufA, N);
        node_scores_kernel<<<cdiv(N, wavesPerBlk), T, 0, stream>>>(
            bufA, a_src[l], a_dst[l], asrc, adst, mkey, denom, N);
        fill_f32<<<cdiv((long)N64, T), T, 0, stream>>>(bufB, 0.f, (int)N64);
        edge_max_kernel<<<cdiv(E + N, T), T, 0, stream>>>(
            ei, ea, sloop, cnt, asrc, adst, wd, pbuf, mkey, E, N);
        edge_exp_kernel<<<cdiv(E + N, T), T, 0, stream>>>(ei, mkey, pbuf, denom, E, N);
        aggregate_kernel<<<cdiv(E + N, wavesPerBlk), T, 0, stream>>>(
            ei, bufA, pbuf, denom, bufB, E, N);
        bias_act_kernel<<<cdiv((long)N64, T), T, 0, stream>>>(
            bufB, bias[l], (int)N64, l == 0 ? 1 : 0);
    }

    pool_kernel<<<cdiv(N, wavesPerBlk), T, 0, stream>>>(bufB, batch, pooled, gcnt, N);
    head_kernel<<<G, 32, 0, stream>>>(pooled, gcnt, fcW, fcb, out, G);
}